// SparseAutoencoder_80272938762630
// MI455X (gfx1250) — compile-verified
//
#include <hip/hip_runtime.h>
#include <hip/hip_bf16.h>
#include <float.h>

// Sparse autoencoder forward for MI455X (gfx1250), wave32.
// out = [ z_hat (N*D) | h_sparse (N*H) | loss_recon, loss_sparse, loss ]

typedef __attribute__((ext_vector_type(2))) float v2f;
typedef __attribute__((ext_vector_type(8))) float v8f;
typedef __attribute__((ext_vector_type(4))) int   v4i;
typedef __attribute__((address_space(1))) v4i*    gptr_v4i;  // global
typedef __attribute__((address_space(3))) v4i*    lptr_v4i;  // LDS

constexpr int   N_ = 16384;
constexpr int   D_ = 512;
constexpr int   H_ = 4096;
constexpr int   K_ = 32;
constexpr float LAMBDA_ = 0.001f;

// Async direct global->LDS copies (CDNA5, ASYNCcnt-tracked), with fallback.
#if defined(__has_builtin)
#if __has_builtin(__builtin_amdgcn_global_load_async_to_lds_b128)
#define HAVE_ASYNC_LDS 1
#endif
#endif

__device__ __forceinline__ void wait_asynccnt0() {
#if defined(__has_builtin) && __has_builtin(__builtin_amdgcn_s_wait_asynccnt)
  __builtin_amdgcn_s_wait_asynccnt(0);
#else
  asm volatile("s_wait_asynccnt 0x0" ::: "memory");
#endif
}

// ---------------------------------------------------------------------------
// Encoder GEMM: h[n,j] = sum_d z[n,d] * W_enc[j,d] + b_enc[j]
// fp32 WMMA 16x16x4. Block = 256 threads (8 waves) -> 64(M) x 128(N) tile.
// Wave w: M-strip (w%4)*16, N-strips (w/4)*64 .. +63 (four 16x16 accumulators).
// LDS row pad = +4 floats: 144B row stride keeps B128 stores 16B-aligned and
// 36*n mod 64 gives conflict-free banks for fragment reads.
// ---------------------------------------------------------------------------
__global__ __launch_bounds__(256) void enc_gemm_kernel(
    const float* __restrict__ z, const float* __restrict__ We,
    const float* __restrict__ be, float* __restrict__ h) {
  __shared__ float As[64][36];    // [m][k]
  __shared__ float Bs[128][36];   // [n][k] == W_enc row-major slice == B^T

  const int t    = threadIdx.x;
  const int lane = t & 31;
  const int wave = t >> 5;
  const int m0 = blockIdx.y * 64;
  const int h0 = blockIdx.x * 128;
  const int mw = (wave & 3) * 16;   // wave M offset in tile
  const int nw = (wave >> 2) * 64;  // wave N offset in tile

  const int lm  = lane & 15;        // M (A) / N (B) within fragment
  const int klo = (lane >> 4) * 2;  // ISA A-layout: lanes 0-15 -> K={0,1}, 16-31 -> K={2,3}

  // float4-granular staging indices: idx4 -> row = idx4/8, col = (idx4%8)*4
  const int arow = t >> 3;
  const int acol = (t & 7) * 4;

  v8f c0 = {}; v8f c1 = {}; v8f c2 = {}; v8f c3 = {};

  for (int k0 = 0; k0 < D_; k0 += 32) {
#if defined(HAVE_ASYNC_LDS)
    // A: 64x32 floats = 512 float4, 2 per thread. B: 128x32 = 1024 float4, 4/thread.
#pragma unroll
    for (int i = 0; i < 2; ++i) {
      int row = arow + i * 32;
      __builtin_amdgcn_global_load_async_to_lds_b128(
          (gptr_v4i)(z + (size_t)(m0 + row) * D_ + k0 + acol),
          (lptr_v4i)(&As[row][acol]), 0, 0);
    }
#pragma unroll
    for (int i = 0; i < 4; ++i) {
      int row = arow + i * 32;
      __builtin_amdgcn_global_load_async_to_lds_b128(
          (gptr_v4i)(We + (size_t)(h0 + row) * D_ + k0 + acol),
          (lptr_v4i)(&Bs[row][acol]), 0, 0);
    }
    wait_asynccnt0();
#else
#pragma unroll
    for (int i = 0; i < 2; ++i) {
      int row = arow + i * 32;
      *(float4*)(&As[row][acol]) =
          *(const float4*)(z + (size_t)(m0 + row) * D_ + k0 + acol);
    }
#pragma unroll
    for (int i = 0; i < 4; ++i) {
      int row = arow + i * 32;
      *(float4*)(&Bs[row][acol]) =
          *(const float4*)(We + (size_t)(h0 + row) * D_ + k0 + acol);
    }
#endif
    __syncthreads();

#pragma unroll
    for (int kk = 0; kk < 32; kk += 4) {
      v2f a, b0, b1, b2, b3;
      a[0]  = As[mw + lm][kk + klo];
      a[1]  = As[mw + lm][kk + klo + 1];
      b0[0] = Bs[nw + lm][kk + klo];
      b0[1] = Bs[nw + lm][kk + klo + 1];
      b1[0] = Bs[nw + 16 + lm][kk + klo];
      b1[1] = Bs[nw + 16 + lm][kk + klo + 1];
      b2[0] = Bs[nw + 32 + lm][kk + klo];
      b2[1] = Bs[nw + 32 + lm][kk + klo + 1];
      b3[0] = Bs[nw + 48 + lm][kk + klo];
      b3[1] = Bs[nw + 48 + lm][kk + klo + 1];
      c0 = __builtin_amdgcn_wmma_f32_16x16x4_f32(false, a, false, b0, (short)0, c0, false, false);
      c1 = __builtin_amdgcn_wmma_f32_16x16x4_f32(false, a, false, b1, (short)0, c1, false, false);
      c2 = __builtin_amdgcn_wmma_f32_16x16x4_f32(false, a, false, b2, (short)0, c2, false, false);
      c3 = __builtin_amdgcn_wmma_f32_16x16x4_f32(false, a, false, b3, (short)0, c3, false, false);
    }
    __syncthreads();
  }

  // C/D layout: VGPR r -> lanes 0-15: M=r, lanes 16-31: M=r+8; N = lane%16.
  const int mbase = m0 + mw + (lane >> 4) * 8;
  v8f acc[4] = {c0, c1, c2, c3};
#pragma unroll
  for (int s = 0; s < 4; ++s) {
    const int n = h0 + nw + s * 16 + lm;
    const float bias = be[n];
#pragma unroll
    for (int r = 0; r < 8; ++r)
      h[(size_t)(mbase + r) * H_ + n] = acc[s][r] + bias;
  }
}

// ---------------------------------------------------------------------------
// Per-row TopK(32) + ReLU, in place: dense h -> h_sparse. One block per row.
// Iterative argmax over an LDS copy (deterministic, ties -> lowest index).
// ---------------------------------------------------------------------------
__global__ __launch_bounds__(256) void topk_kernel(
    float* __restrict__ h, int* __restrict__ ws_idx,
    float* __restrict__ ws_val, float* __restrict__ partial_sparse) {
  __shared__ float vals[H_];      // 16 KB row copy
  __shared__ float rv[256];
  __shared__ int   ri[256];
  __shared__ int   sel_idx[K_];
  __shared__ float sel_val[K_];

  const int row = blockIdx.x;
  const int t   = threadIdx.x;
  float* hrow = h + (size_t)row * H_;

  for (int i = t; i < H_; i += 256) vals[i] = hrow[i];
  __syncthreads();

  for (int k = 0; k < K_; ++k) {
    float bv = -FLT_MAX; int bi = 0;
    for (int i = t; i < H_; i += 256) {
      float v = vals[i];
      if (v > bv) { bv = v; bi = i; }   // strict > keeps lowest index on ties
    }
    rv[t] = bv; ri[t] = bi;
    __syncthreads();
    for (int s = 128; s > 0; s >>= 1) {
      if (t < s) {
        float v2 = rv[t + s]; int i2 = ri[t + s];
        if (v2 > rv[t] || (v2 == rv[t] && i2 < ri[t])) { rv[t] = v2; ri[t] = i2; }
      }
      __syncthreads();
    }
    if (t == 0) {
      sel_idx[k] = ri[0];
      sel_val[k] = rv[0];
      vals[ri[0]] = -FLT_MAX;  // remove from further selection
    }
    __syncthreads();
  }

  // Rewrite row: zeros everywhere, relu(val) at selected indices.
  for (int i = t; i < H_; i += 256) hrow[i] = 0.0f;
  __syncthreads();
  if (t < K_) {
    float v = sel_val[t];
    v = v > 0.0f ? v : 0.0f;
    hrow[sel_idx[t]]     = v;
    ws_idx[row * K_ + t] = sel_idx[t];
    ws_val[row * K_ + t] = v;
  }
  if (t == 0) {
    float s = 0.0f;
#pragma unroll
    for (int k = 0; k < K_; ++k) { float v = sel_val[k]; s += (v > 0.0f ? v : 0.0f); }
    partial_sparse[row] = s;
  }
}

// ---------------------------------------------------------------------------
// W_dec [D,H] -> Wt [H,D] transpose (8 MB, L2-resident afterwards).
// ---------------------------------------------------------------------------
__global__ __launch_bounds__(256) void transpose_wdec_kernel(
    const float* __restrict__ Wd, float* __restrict__ Wt) {
  __shared__ float tile[32][33];
  const int t  = threadIdx.x;
  const int tx = t & 31, ty = t >> 5;
  const int hb = blockIdx.x * 32;   // H tile
  const int db = blockIdx.y * 32;   // D tile
#pragma unroll
  for (int i = 0; i < 4; ++i)
    tile[ty + i * 8][tx] = Wd[(size_t)(db + ty + i * 8) * H_ + hb + tx];
  __syncthreads();
#pragma unroll
  for (int i = 0; i < 4; ++i)
    Wt[(size_t)(hb + ty + i * 8) * D_ + db + tx] = tile[tx][ty + i * 8];
}

// ---------------------------------------------------------------------------
// Sparse decoder + recon-loss partials. One block per sample row:
// z_hat[n,:] = b_dec + sum_k val_k * Wt[idx_k, :]   (32 coalesced row gathers)
// ---------------------------------------------------------------------------
__global__ __launch_bounds__(256) void dec_kernel(
    const float* __restrict__ z, const float* __restrict__ Wt,
    const float* __restrict__ bd, const int* __restrict__ ws_idx,
    const float* __restrict__ ws_val, float* __restrict__ zhat,
    float* __restrict__ partial_recon) {
  __shared__ int   s_idx[K_];
  __shared__ float s_val[K_];
  __shared__ float red[256];

  const int row = blockIdx.x;
  const int t   = threadIdx.x;
  if (t < K_) { s_idx[t] = ws_idx[row * K_ + t]; s_val[t] = ws_val[row * K_ + t]; }
  __syncthreads();

  float acc0 = bd[t];
  float acc1 = bd[t + 256];
#pragma unroll 4
  for (int k = 0; k < K_; ++k) {
    const float* wrow = Wt + (size_t)s_idx[k] * D_;
    const float v = s_val[k];
    acc0 += v * wrow[t];
    acc1 += v * wrow[t + 256];
  }
  const size_t base = (size_t)row * D_;
  zhat[base + t]       = acc0;
  zhat[base + t + 256] = acc1;

  const float e0 = acc0 - z[base + t];
  const float e1 = acc1 - z[base + t + 256];
  red[t] = e0 * e0 + e1 * e1;
  __syncthreads();
  for (int s = 128; s > 0; s >>= 1) {
    if (t < s) red[t] += red[t + s];
    __syncthreads();
  }
  if (t == 0) partial_recon[row] = red[0];
}

// ---------------------------------------------------------------------------
// Final deterministic reduction of per-row partials -> 3 scalar losses.
// ---------------------------------------------------------------------------
__global__ __launch_bounds__(256) void loss_kernel(
    const float* __restrict__ partial_recon,
    const float* __restrict__ partial_sparse, float* __restrict__ out_losses) {
  __shared__ float r1[256], r2[256];
  const int t = threadIdx.x;
  float a = 0.0f, b = 0.0f;
  for (int i = t; i < N_; i += 256) { a += partial_recon[i]; b += partial_sparse[i]; }
  r1[t] = a; r2[t] = b;
  __syncthreads();
  for (int s = 128; s > 0; s >>= 1) {
    if (t < s) { r1[t] += r1[t + s]; r2[t] += r2[t + s]; }
    __syncthreads();
  }
  if (t == 0) {
    const float lr = r1[0] * (1.0f / ((float)N_ * (float)D_));
    const float ls = r2[0] * (1.0f / ((float)N_ * (float)H_));
    out_losses[0] = lr;
    out_losses[1] = ls;
    out_losses[2] = lr + LAMBDA_ * ls;
  }
}

// ---------------------------------------------------------------------------
extern "C" void kernel_launch(void* const* d_in, const int* in_sizes, int n_in,
                              void* d_out, int out_size, void* d_ws, size_t ws_size,
                              hipStream_t stream) {
  const float* z  = (const float*)d_in[0];   // [N, D]
  const float* We = (const float*)d_in[1];   // [H, D]
  const float* be = (const float*)d_in[2];   // [H]
  const float* Wd = (const float*)d_in[3];   // [D, H]
  const float* bd = (const float*)d_in[4];   // [D]

  float* zhat   = (float*)d_out;                       // N*D
  float* h      = zhat + (size_t)N_ * D_;              // N*H (dense h, then h_sparse in place)
  float* losses = h + (size_t)N_ * H_;                 // 3 scalars

  char* ws = (char*)d_ws;
  float* Wt       = (float*)ws; ws += sizeof(float) * (size_t)H_ * D_;  // 8 MB
  int*   ws_idx   = (int*)  ws; ws += sizeof(int)   * (size_t)N_ * K_;  // 2 MB
  float* ws_val   = (float*)ws; ws += sizeof(float) * (size_t)N_ * K_;  // 2 MB
  float* p_recon  = (float*)ws; ws += sizeof(float) * (size_t)N_;       // 64 KB
  float* p_sparse = (float*)ws;                                         // 64 KB

  transpose_wdec_kernel<<<dim3(H_ / 32, D_ / 32), 256, 0, stream>>>(Wd, Wt);
  enc_gemm_kernel<<<dim3(H_ / 128, N_ / 64), 256, 0, stream>>>(z, We, be, h);
  topk_kernel<<<N_, 256, 0, stream>>>(h, ws_idx, ws_val, p_sparse);
  dec_kernel<<<N_, 256, 0, stream>>>(z, Wt, bd, ws_idx, ws_val, zhat, p_recon);
  loss_kernel<<<1, 256, 0, stream>>>(p_recon, p_sparse, losses);
}